// ManualHGTConv_29738353558363
// MI455X (gfx1250) — compile-verified
//
#include <hip/hip_runtime.h>
#include <hip/hip_bf16.h>

#define DEVINL __device__ __forceinline__

typedef float v2f __attribute__((ext_vector_type(2)));
typedef float v8f __attribute__((ext_vector_type(8)));

constexpr int N_NODES = 30000;
constexpr int IN_DIM  = 256;
constexpr int OUT_DIM = 256;
constexpr int H       = 4;
constexpr int HD      = 64;          // head dim
constexpr int T_TYPES = 5;
constexpr int E_PER   = 40000;
constexpr int TE      = T_TYPES * E_PER;   // 200000
constexpr int KQV_LD  = 3 * OUT_DIM;       // 768

DEVINL float gelu_exact(float x) {
    return 0.5f * x * (1.0f + erff(x * 0.70710678118654752440f));
}

// monotonic float <-> uint mapping so unsigned atomicMax == float max
DEVINL unsigned ordf(float f) {
    unsigned u = __float_as_uint(f);
    return (u & 0x80000000u) ? ~u : (u | 0x80000000u);
}
DEVINL float unordf(unsigned u) {
    return __uint_as_float((u & 0x80000000u) ? (u & 0x7FFFFFFFu) : ~u);
}

// One-wave 16x16 output tile of D = op(A) * B using V_WMMA_F32_16X16X4_F32.
// A(m,k) = A[m*ldA + k]   (row-major tile, 16 x K)
// B(k,n) = B[k*sBk + n*sBn]  (strides cover both W-transposed and plain layouts)
// ISA lane layout (wave32):
//   A 16x4: lane l (l<16): M=l, K={k0,k0+1}; lane l+16: M=l, K={k0+2,k0+3}
//   B 4x16: lane l (l<16): N=l, K={k0,k0+1}; lane l+16: N=l, K={k0+2,k0+3}
//   C/D   : vgpr j -> M = j + 8*(lane>=16), N = lane&15
template <bool GELU_A>
DEVINL v8f wmma_tile_f32(const float* __restrict__ A, int ldA,
                         const float* __restrict__ B, int sBk, int sBn, int K)
{
    const int lane = threadIdx.x & 31;
    const int half = lane >> 4;
    const int l    = lane & 15;
    const float* arow = A + (size_t)l * ldA;
    v8f acc = {};
    for (int k0 = 0; k0 < K; k0 += 4) {
        const int ka = k0 + 2 * half;
        float a0 = arow[ka];
        float a1 = arow[ka + 1];
        if (GELU_A) { a0 = gelu_exact(a0); a1 = gelu_exact(a1); }
        v2f a; a[0] = a0; a[1] = a1;
        v2f b;
        b[0] = B[(size_t)ka * sBk + (size_t)l * sBn];
        b[1] = B[(size_t)(ka + 1) * sBk + (size_t)l * sBn];
        acc = __builtin_amdgcn_wmma_f32_16x16x4_f32(
                  false, a, false, b, (short)0, acc, false, false);
    }
    return acc;
}

// ---------------- kernels ----------------

// zero amax (encoded -inf == 0), expsum, out accumulator
__global__ void init_ws_kernel(unsigned* __restrict__ amax,
                               float* __restrict__ expsum,
                               float* __restrict__ outacc)
{
    const int i = blockIdx.x * blockDim.x + threadIdx.x;   // total = N*OUT_DIM
    if (i < N_NODES * H) { amax[i] = 0u; expsum[i] = 0.0f; }
    outacc[i] = 0.0f;
}

// kqv[N,768] = h[N,256] @ W[768,256]^T + bias
__global__ void __launch_bounds__(256)
kqv_gemm_kernel(const float* __restrict__ h, const float* __restrict__ W,
                const float* __restrict__ bias, float* __restrict__ kqv)
{
    const int wave = threadIdx.x >> 5;                 // 8 waves
    const int m0 = blockIdx.x * 16;
    const int n0 = (blockIdx.y * 8 + wave) * 16;
    // B(k,n) = W[(n0+n)*IN_DIM + k]
    v8f c = wmma_tile_f32<false>(h + (size_t)m0 * IN_DIM, IN_DIM,
                                 W + (size_t)n0 * IN_DIM, 1, IN_DIM, IN_DIM);
    const int lane = threadIdx.x & 31;
    const int half = lane >> 4, l = lane & 15;
    const int col = n0 + l;
    const float bv = bias[col];
#pragma unroll
    for (int j = 0; j < 8; ++j) {
        const int row = m0 + j + 8 * half;
        kqv[(size_t)row * KQV_LD + col] = c[j] + bv;
    }
}

// Kt[t][n][h*64+f] = sum_d K[n,h,d] * k_rel[t][d][f]; same for Vt with v_rel.
// blockIdx.y encodes (which, t, head); 4 waves cover the 4 f-tiles of 64.
__global__ void __launch_bounds__(128)
rel_gemm_kernel(const float* __restrict__ kqv,
                const float* __restrict__ k_rel, const float* __restrict__ v_rel,
                float* __restrict__ Kt, float* __restrict__ Vt)
{
    const int wave = threadIdx.x >> 5;                 // f-tile 0..3
    const int z = blockIdx.y;                          // 0..39
    const int which = z / (T_TYPES * H);
    const int rem   = z % (T_TYPES * H);
    const int t  = rem / H;
    const int hh = rem % H;
    const float* rel = (which ? v_rel : k_rel) + (size_t)t * HD * HD;
    const float* Ab  = kqv + (size_t)blockIdx.x * 16 * KQV_LD
                           + (which ? 2 * OUT_DIM : 0) + hh * HD;
    float* Ob = (which ? Vt : Kt)
              + ((size_t)t * N_NODES + (size_t)blockIdx.x * 16) * OUT_DIM + hh * HD;
    const int f0 = wave * 16;
    // B(k,f) = rel[k*64 + f0+f]
    v8f c = wmma_tile_f32<false>(Ab, KQV_LD, rel + f0, HD, 1, HD);
    const int lane = threadIdx.x & 31;
    const int half = lane >> 4, l = lane & 15;
#pragma unroll
    for (int j = 0; j < 8; ++j) {
        const int rl = j + 8 * half;
        Ob[(size_t)rl * OUT_DIM + f0 + l] = c[j];
    }
}

// alpha[t,e,h] = scale * p_rel[t,h] * dot(Q[tgt,h,:], Kt[t,src,h,:]); seg-max
__global__ void __launch_bounds__(256)
alpha_max_kernel(const int* __restrict__ esrc, const int* __restrict__ etgt,
                 const float* __restrict__ kqv, const float* __restrict__ Kt,
                 const float* __restrict__ p_rel,
                 float* __restrict__ alpha, unsigned* __restrict__ amax)
{
    const int idx = blockIdx.x * blockDim.x + threadIdx.x;   // TE*H threads
    const int eg = idx >> 2;
    const int hh = idx & 3;
    const int t  = eg / E_PER;
    const int src = esrc[eg];
    const int tgt = etgt[eg];
    const float4* q = (const float4*)(kqv + (size_t)tgt * KQV_LD + OUT_DIM + hh * HD);
    const float4* k = (const float4*)(Kt + ((size_t)t * N_NODES + src) * OUT_DIM + hh * HD);
    float s = 0.0f;
#pragma unroll
    for (int i = 0; i < 16; ++i) {
        const float4 qa = q[i], kb = k[i];
        s += qa.x * kb.x + qa.y * kb.y + qa.z * kb.z + qa.w * kb.w;
    }
    const float av = s * 0.125f * p_rel[t * H + hh];   // 64^-0.5 = 0.125
    alpha[idx] = av;
    atomicMax(&amax[(size_t)tgt * H + hh], ordf(av));
}

// exp(alpha - segmax) and segmented sum
__global__ void __launch_bounds__(256)
exp_sum_kernel(const int* __restrict__ etgt, float* __restrict__ alpha,
               const unsigned* __restrict__ amax, float* __restrict__ expsum)
{
    const int idx = blockIdx.x * blockDim.x + threadIdx.x;
    const int eg = idx >> 2;
    const int hh = idx & 3;
    const int tgt = etgt[eg];
    const float m = unordf(amax[(size_t)tgt * H + hh]);
    const float ea = expf(alpha[idx] - m);
    alpha[idx] = ea;   // overwrite with exp value (recomputed fresh each call)
    atomicAdd(&expsum[(size_t)tgt * H + hh], ea);
}

// out[tgt,h,:] += attn * Vt[t,src,h,:]
__global__ void __launch_bounds__(256)
scatter_kernel(const int* __restrict__ esrc, const int* __restrict__ etgt,
               const float* __restrict__ alpha, const float* __restrict__ expsum,
               const float* __restrict__ Vt, float* __restrict__ outacc)
{
    const int idx = blockIdx.x * blockDim.x + threadIdx.x;
    const int eg = idx >> 2;
    const int hh = idx & 3;
    const int t  = eg / E_PER;
    const int src = esrc[eg];
    const int tgt = etgt[eg];
    const float attn = alpha[idx] / (expsum[(size_t)tgt * H + hh] + 1e-16f);
    const float4* v = (const float4*)(Vt + ((size_t)t * N_NODES + src) * OUT_DIM + hh * HD);
    float* o = outacc + (size_t)tgt * OUT_DIM + hh * HD;
#pragma unroll
    for (int i = 0; i < 16; ++i) {
        const float4 vv = v[i];
        atomicAdd(o + 4 * i + 0, attn * vv.x);
        atomicAdd(o + 4 * i + 1, attn * vv.y);
        atomicAdd(o + 4 * i + 2, attn * vv.z);
        atomicAdd(o + 4 * i + 3, attn * vv.w);
    }
}

// out = a * (gelu(acc) @ outW^T + outB) + (1-a) * h
__global__ void __launch_bounds__(256)
out_gemm_kernel(const float* __restrict__ outacc, const float* __restrict__ W,
                const float* __restrict__ bias, const float* __restrict__ h,
                const float* __restrict__ skip, float* __restrict__ out)
{
    const int wave = threadIdx.x >> 5;                 // 8 waves
    const int m0 = blockIdx.x * 16;
    const int n0 = (blockIdx.y * 8 + wave) * 16;
    v8f c = wmma_tile_f32<true>(outacc + (size_t)m0 * OUT_DIM, OUT_DIM,
                                W + (size_t)n0 * OUT_DIM, 1, OUT_DIM, OUT_DIM);
    const float aa = 1.0f / (1.0f + expf(-skip[0]));
    const int lane = threadIdx.x & 31;
    const int half = lane >> 4, l = lane & 15;
    const int col = n0 + l;
    const float bv = bias[col];
#pragma unroll
    for (int j = 0; j < 8; ++j) {
        const int row = m0 + j + 8 * half;
        const size_t p = (size_t)row * OUT_DIM + col;
        out[p] = aa * (c[j] + bv) + (1.0f - aa) * h[p];
    }
}

// ---------------- launch ----------------

extern "C" void kernel_launch(void* const* d_in, const int* in_sizes, int n_in,
                              void* d_out, int out_size, void* d_ws, size_t ws_size,
                              hipStream_t stream)
{
    const float* h     = (const float*)d_in[0];
    const int*   esrc  = (const int*)d_in[1];
    const int*   etgt  = (const int*)d_in[2];
    const float* kqvW  = (const float*)d_in[3];
    const float* kqvB  = (const float*)d_in[4];
    const float* outW  = (const float*)d_in[5];
    const float* outB  = (const float*)d_in[6];
    const float* k_rel = (const float*)d_in[7];
    const float* v_rel = (const float*)d_in[8];
    const float* skip  = (const float*)d_in[9];
    const float* p_rel = (const float*)d_in[10];
    float* out = (float*)d_out;

    float* ws = (float*)d_ws;
    size_t off = 0;
    float*    kqv    = ws + off; off += (size_t)N_NODES * KQV_LD;            // 23.04M
    float*    Kt     = ws + off; off += (size_t)T_TYPES * N_NODES * OUT_DIM; // 38.4M
    float*    Vt     = ws + off; off += (size_t)T_TYPES * N_NODES * OUT_DIM; // 38.4M
    float*    alpha  = ws + off; off += (size_t)TE * H;                      // 0.8M
    unsigned* amax   = (unsigned*)(ws + off); off += (size_t)N_NODES * H;
    float*    expsum = ws + off; off += (size_t)N_NODES * H;
    float*    outacc = ws + off; off += (size_t)N_NODES * OUT_DIM;           // 7.68M

    init_ws_kernel<<<(N_NODES * OUT_DIM) / 256, 256, 0, stream>>>(amax, expsum, outacc);
    kqv_gemm_kernel<<<dim3(N_NODES / 16, KQV_LD / 128), 256, 0, stream>>>(h, kqvW, kqvB, kqv);
    rel_gemm_kernel<<<dim3(N_NODES / 16, 2 * T_TYPES * H), 128, 0, stream>>>(kqv, k_rel, v_rel, Kt, Vt);
    alpha_max_kernel<<<(TE * H) / 256, 256, 0, stream>>>(esrc, etgt, kqv, Kt, p_rel, alpha, amax);
    exp_sum_kernel<<<(TE * H) / 256, 256, 0, stream>>>(etgt, alpha, amax, expsum);
    scatter_kernel<<<(TE * H) / 256, 256, 0, stream>>>(esrc, etgt, alpha, expsum, Vt, outacc);
    out_gemm_kernel<<<dim3(N_NODES / 16, OUT_DIM / 128), 256, 0, stream>>>(outacc, outW, outB, h, skip, out);
}